// LigandGINE_1254130450544
// MI455X (gfx1250) — compile-verified
//
#include <hip/hip_runtime.h>
#include <math.h>

typedef __attribute__((ext_vector_type(2))) float v2f;
typedef __attribute__((ext_vector_type(8))) float v8f;

#define NUM_LAYERS 3
#define H 128
#define EDGE_CH 4
#define N_NODES 100000
#define N_EDGES 600000
#define TILE_M 16
#define WAVES_PER_BLOCK 5         // 6250 tiles / 5 = 1250 blocks, exact
#define LDS_STRIDE 132            // 128 + 4 pad: removes LDS bank conflicts, keeps 16B align

// ---------------------------------------------------------------------------
// Kernel 1: x[i] = emb[z[i]]   (wave per node, float4 per lane)
// ---------------------------------------------------------------------------
__global__ __launch_bounds__(256) void gine_embed(const float* __restrict__ emb,
                                                  const int* __restrict__ z,
                                                  float* __restrict__ x) {
    int t    = blockIdx.x * 256 + threadIdx.x;
    int node = t >> 5;
    int lane = t & 31;
    if (node >= N_NODES) return;
    int c = lane * 4;
    const float4 v = *(const float4*)(emb + (size_t)z[node] * H + c);
    *(float4*)(x + (size_t)node * H + c) = v;
}

// ---------------------------------------------------------------------------
// Kernel 2: per-edge fused: e = attr@We + be ; m = relu(x[src]+e) ;
//           atomic scatter-add into agg[dst].  Wave per edge, 4 ch per lane.
// ---------------------------------------------------------------------------
__global__ __launch_bounds__(256) void gine_edge(const float* __restrict__ x,
                                                 const float* __restrict__ edge_attr,
                                                 const float* __restrict__ Wel,  // [4][128]
                                                 const float* __restrict__ bel,  // [128]
                                                 const int* __restrict__ ei,     // [2][E]
                                                 float* __restrict__ agg) {
    int t    = blockIdx.x * 256 + threadIdx.x;
    int e    = t >> 5;
    int lane = t & 31;
    if (e >= N_EDGES) return;

    int src = ei[e];
    int dst = ei[N_EDGES + e];
    // EDGE_CH == 4: one aligned float4, same address across the wave -> broadcast
    const float4 a = *(const float4*)(edge_attr + (size_t)e * EDGE_CH);

    int c = lane * 4;
    const float4 w0 = *(const float4*)(Wel + 0 * H + c);
    const float4 w1 = *(const float4*)(Wel + 1 * H + c);
    const float4 w2 = *(const float4*)(Wel + 2 * H + c);
    const float4 w3 = *(const float4*)(Wel + 3 * H + c);
    const float4 bb = *(const float4*)(bel + c);

    float ex = bb.x + a.x * w0.x + a.y * w1.x + a.z * w2.x + a.w * w3.x;
    float ey = bb.y + a.x * w0.y + a.y * w1.y + a.z * w2.y + a.w * w3.y;
    float ez = bb.z + a.x * w0.z + a.y * w1.z + a.z * w2.z + a.w * w3.z;
    float ew = bb.w + a.x * w0.w + a.y * w1.w + a.z * w2.w + a.w * w3.w;

    const float4 xs = *(const float4*)(x + (size_t)src * H + c);
    float mx = fmaxf(xs.x + ex, 0.0f);
    float my = fmaxf(xs.y + ey, 0.0f);
    float mz = fmaxf(xs.z + ez, 0.0f);
    float mw = fmaxf(xs.w + ew, 0.0f);

    float* ap = agg + (size_t)dst * H + c;
    atomicAdd(ap + 0, mx);
    atomicAdd(ap + 1, my);
    atomicAdd(ap + 2, mz);
    atomicAdd(ap + 3, mw);
}

// ---------------------------------------------------------------------------
// Kernel 3: fused MLP per 16-node tile, full fp32 via V_WMMA_F32_16X16X4_F32.
//   h = x + agg ; t = silu(h@W1 + b1) ; x = t@W2 + b2      (in place)
// One wave per tile; 5 waves (tiles) per block; 6250 tiles total (exact).
// ---------------------------------------------------------------------------
__global__ __launch_bounds__(32 * WAVES_PER_BLOCK) void gine_mlp(
        float* __restrict__ x,               // [N][H] read+write (in place)
        const float* __restrict__ agg,       // [N][H]
        const float* __restrict__ W1l,       // [H][H] row-major (K x N)
        const float* __restrict__ b1l,       // [H]
        const float* __restrict__ W2l,       // [H][H]
        const float* __restrict__ b2l,       // [H]
        float* __restrict__ out) {           // [N][H] (== x in every layer)
    __shared__ float hbuf[WAVES_PER_BLOCK * TILE_M * LDS_STRIDE];

    const int lane = threadIdx.x & 31;
    const int wave = threadIdx.x >> 5;
    const int tile = blockIdx.x * WAVES_PER_BLOCK + wave;
    const int row0 = tile * TILE_M;

    float* hb = hbuf + wave * (TILE_M * LDS_STRIDE);
    const int m    = lane & 15;    // A-row / C-column base
    const int half = lane >> 4;    // lane half selects K-pair / M-upper
    const int n0   = lane & 15;

    // ---- stage h = x + agg into LDS (row-major, padded stride) ----
    #pragma unroll 4
    for (int r = 0; r < TILE_M; ++r) {
        const size_t g = (size_t)(row0 + r) * H + lane * 4;
        const float4 xv = *(const float4*)(x + g);
        const float4 av = *(const float4*)(agg + g);
        float4 h4;
        h4.x = xv.x + av.x; h4.y = xv.y + av.y;
        h4.z = xv.z + av.z; h4.w = xv.w + av.w;
        *(float4*)(hb + r * LDS_STRIDE + lane * 4) = h4;
    }
    __syncthreads();

    // ---- GEMM1: acc = h @ W1 + b1 ----
    v8f acc[8];
    #pragma unroll
    for (int nt = 0; nt < 8; ++nt) {
        const float bv = b1l[nt * 16 + n0];
        #pragma unroll
        for (int i = 0; i < 8; ++i) acc[nt][i] = bv;
    }
    for (int kb = 0; kb < H; kb += 4) {
        // A fragment 16x4: lane half selects K pair {kb+2h, kb+2h+1}
        const v2f a = *(const v2f*)(hb + m * LDS_STRIDE + kb + 2 * half);
        const float* wr0 = W1l + (size_t)(kb + 2 * half) * H;
        #pragma unroll
        for (int nt = 0; nt < 8; ++nt) {
            v2f b;
            b.x = wr0[nt * 16 + n0];
            b.y = wr0[H + nt * 16 + n0];
            acc[nt] = __builtin_amdgcn_wmma_f32_16x16x4_f32(
                false, a, false, b, (short)0, acc[nt], false, false);
        }
    }

    // ---- silu + C-layout -> A-layout via LDS ----
    __syncthreads();   // all waves done reading hbuf before overwrite
    #pragma unroll
    for (int nt = 0; nt < 8; ++nt) {
        #pragma unroll
        for (int r = 0; r < 8; ++r) {
            float v = acc[nt][r];
            v = v / (1.0f + expf(-v));               // silu
            hb[(r + 8 * half) * LDS_STRIDE + nt * 16 + n0] = v;
        }
    }
    __syncthreads();

    // ---- GEMM2: acc = t @ W2 + b2 ----
    #pragma unroll
    for (int nt = 0; nt < 8; ++nt) {
        const float bv = b2l[nt * 16 + n0];
        #pragma unroll
        for (int i = 0; i < 8; ++i) acc[nt][i] = bv;
    }
    for (int kb = 0; kb < H; kb += 4) {
        const v2f a = *(const v2f*)(hb + m * LDS_STRIDE + kb + 2 * half);
        const float* wr0 = W2l + (size_t)(kb + 2 * half) * H;
        #pragma unroll
        for (int nt = 0; nt < 8; ++nt) {
            v2f b;
            b.x = wr0[nt * 16 + n0];
            b.y = wr0[H + nt * 16 + n0];
            acc[nt] = __builtin_amdgcn_wmma_f32_16x16x4_f32(
                false, a, false, b, (short)0, acc[nt], false, false);
        }
    }

    // ---- write result (C layout -> row-major global) ----
    #pragma unroll
    for (int nt = 0; nt < 8; ++nt) {
        #pragma unroll
        for (int r = 0; r < 8; ++r) {
            out[(size_t)(row0 + r + 8 * half) * H + nt * 16 + n0] = acc[nt][r];
        }
    }
}

// ---------------------------------------------------------------------------
// Kernel 4: pass-through batch_vec as float into the output tail
// ---------------------------------------------------------------------------
__global__ __launch_bounds__(256) void gine_batch(const int* __restrict__ bvec,
                                                  float* __restrict__ outTail) {
    int i = blockIdx.x * 256 + threadIdx.x;
    if (i < N_NODES) outTail[i] = (float)bvec[i];
}

// ---------------------------------------------------------------------------
extern "C" void kernel_launch(void* const* d_in, const int* in_sizes, int n_in,
                              void* d_out, int out_size, void* d_ws, size_t ws_size,
                              hipStream_t stream) {
    const float* emb       = (const float*)d_in[0];
    const float* We        = (const float*)d_in[1];
    const float* be        = (const float*)d_in[2];
    const float* W1        = (const float*)d_in[3];
    const float* b1        = (const float*)d_in[4];
    const float* W2        = (const float*)d_in[5];
    const float* b2        = (const float*)d_in[6];
    const float* edge_attr = (const float*)d_in[7];
    const int*   z         = (const int*)d_in[8];
    const int*   ei        = (const int*)d_in[9];
    const int*   bvec      = (const int*)d_in[10];

    float* out = (float*)d_out;
    float* x   = out;                 // use output x-region as the working buffer
    float* agg = (float*)d_ws;        // 100000*128*4 = 51.2 MB scratch

    // x = emb[z]
    gine_embed<<<N_NODES / 8, 256, 0, stream>>>(emb, z, x);

    for (int l = 0; l < NUM_LAYERS; ++l) {
        hipMemsetAsync(agg, 0, (size_t)N_NODES * H * sizeof(float), stream);
        gine_edge<<<N_EDGES / 8, 256, 0, stream>>>(
            x, edge_attr, We + (size_t)l * EDGE_CH * H, be + (size_t)l * H, ei, agg);
        gine_mlp<<<(N_NODES / TILE_M) / WAVES_PER_BLOCK, 32 * WAVES_PER_BLOCK, 0, stream>>>(
            x, agg,
            W1 + (size_t)l * H * H, b1 + (size_t)l * H,
            W2 + (size_t)l * H * H, b2 + (size_t)l * H,
            x);
    }

    gine_batch<<<(N_NODES + 255) / 256, 256, 0, stream>>>(bvec, out + (size_t)N_NODES * H);
}